// GCNBlock_35210141893223
// MI455X (gfx1250) — compile-verified
//
#include <hip/hip_runtime.h>
#include <hip/hip_bf16.h>
#include <math.h>

#define N_NODES 50000
#define FDIM 64

typedef __attribute__((ext_vector_type(2))) float v2f;
typedef __attribute__((ext_vector_type(8))) float v8f;
typedef __attribute__((ext_vector_type(4))) unsigned int v4u;
typedef __attribute__((ext_vector_type(8))) int v8i;
typedef __attribute__((ext_vector_type(4))) int v4i;

// ---------------- Phase 1: degree + normalization ----------------

__global__ __launch_bounds__(256) void init_deg_kernel(float* __restrict__ deg) {
    int i = blockIdx.x * blockDim.x + threadIdx.x;
    if (i < N_NODES) deg[i] = 1.0f;   // self-loop contributes 1 to in-degree
}

__global__ __launch_bounds__(256) void count_deg_kernel(const long long* __restrict__ dsts,
                                                        float* __restrict__ deg, int E) {
    int e = blockIdx.x * blockDim.x + threadIdx.x;
    if (e < E) unsafeAtomicAdd(&deg[(int)dsts[e]], 1.0f);
}

__global__ __launch_bounds__(256) void dinv_kernel(const float* __restrict__ deg,
                                                   float* __restrict__ dinv) {
    int i = blockIdx.x * blockDim.x + threadIdx.x;
    if (i < N_NODES) dinv[i] = rsqrtf(deg[i]);   // deg >= 1 always (self loops)
}

// ---------------- Phase 2: shared linear aggregation A = scatter(norm * x[src]) ----

// Self-loop term: A[i] = dinv[i]^2 * x[i]
__global__ __launch_bounds__(256) void agg_init_kernel(const float* __restrict__ x,
                                                       const float* __restrict__ dinv,
                                                       float* __restrict__ A) {
    int i = blockIdx.x * blockDim.x + threadIdx.x;   // over N*64
    if (i < N_NODES * FDIM) {
        float di = dinv[i >> 6];
        A[i] = di * di * x[i];
    }
}

// Edge term: one 32-lane wave per edge, each lane handles 2 consecutive features.
__global__ __launch_bounds__(256) void edge_scatter_kernel(const long long* __restrict__ srcs,
                                                           const long long* __restrict__ dsts,
                                                           const float* __restrict__ x,
                                                           const float* __restrict__ dinv,
                                                           float* __restrict__ A, int E) {
    int gid  = blockIdx.x * blockDim.x + threadIdx.x;
    int e    = gid >> 5;
    int lane = gid & 31;
    if (e >= E) return;
    int s = (int)srcs[e];
    int d = (int)dsts[e];
    float norm = dinv[s] * dinv[d];
    const float2 xv = *(const float2*)(x + s * FDIM + lane * 2);
    float* ap = A + d * FDIM + lane * 2;
    unsafeAtomicAdd(ap + 0, norm * xv.x);
    unsafeAtomicAdd(ap + 1, norm * xv.y);
}

// ---------------- Phase 3: dense epilogue with f32 WMMA + TDM-staged weights ----
// out = relu((A@W1)/deg) * sigmoid((A@W2)/deg)
// Grid: 3125 node-tiles (16 nodes) x 4 waves (one 16-wide feature slice each).
// W1,W2 staged to LDS by the Tensor Data Mover (one 64x64 f32 tile each).

__device__ __forceinline__ void tdm_load_w(unsigned int ldsOff, const float* W) {
    const unsigned long long ga = (unsigned long long)(uintptr_t)W;
    v4u g0;
    g0[0] = 1u;                                   // count=1, is_restore=0, gather off
    g0[1] = ldsOff;                               // lds_addr (bytes)
    g0[2] = (unsigned int)(ga & 0xffffffffull);   // global_addr[31:0]
    g0[3] = (unsigned int)((ga >> 32) & 0x01ffffffull) | 0x80000000u; // addr[56:32] | type=2
    v8i g1;
    g1[0] = 0x00020000;        // workgroup_mask=0, data_size=2 (4B), no pad/iterate/barrier
    g1[1] = (FDIM << 16);      // tensor_dim0[15:0]=64 at bits 63:48
    g1[2] = (FDIM << 16);      // tensor_dim0[31:16]=0 ; tensor_dim1[15:0]=64 at bits 95:80
    g1[3] = (FDIM << 16);      // tensor_dim1[31:16]=0 ; tile_dim0=64 at bits 127:112
    g1[4] = FDIM;              // tile_dim1=64 ; tile_dim2=0
    g1[5] = FDIM;              // tensor_dim0_stride[31:0]=64
    g1[6] = 0;                 // stride0[47:32]=0 ; stride1 low=0
    g1[7] = 0;
    v4i z4 = {0, 0, 0, 0};     // groups 2/3 unused for 2D tile
    v8i z8 = {0, 0, 0, 0, 0, 0, 0, 0};
    __builtin_amdgcn_tensor_load_to_lds(g0, g1, z4, z4, z8, 0);
}

__global__ __launch_bounds__(128) void gcn_wmma_kernel(const float* __restrict__ A,
                                                       const float* __restrict__ W1,
                                                       const float* __restrict__ W2,
                                                       const float* __restrict__ deg,
                                                       float* __restrict__ out) {
    __shared__ float ldsW[2 * FDIM * FDIM];      // 32 KB: W1 then W2, row-major

    const int tile  = blockIdx.x;                // node tile
    const int wave  = threadIdx.x >> 5;          // feature slice 0..3
    const int lane  = threadIdx.x & 31;
    const int half  = lane >> 4;                 // 0 or 1
    const int l16   = lane & 15;
    const int node0 = tile * 16;
    const int nBase = wave * 16;

    if (threadIdx.x < 32) {                      // one wave issues both DMAs
        const unsigned int ldsOff = (unsigned int)(uintptr_t)(&ldsW[0]);
        tdm_load_w(ldsOff, W1);
        tdm_load_w(ldsOff + FDIM * FDIM * 4, W2);
        __builtin_amdgcn_s_wait_tensorcnt(0);
    }
    __syncthreads();

    v8f c1 = {};
    v8f c2 = {};

    const float* arow = A + (node0 + l16) * FDIM + half * 2;  // lane's row, K-half offset
    const float* w1l  = ldsW + nBase + l16;
    const float* w2l  = ldsW + FDIM * FDIM + nBase + l16;

#pragma unroll
    for (int kc = 0; kc < 16; ++kc) {
        const int k0 = kc * 4 + half * 2;        // lane's first K of the 4-wide chunk
        v2f a;
        a.x = arow[kc * 4 + 0];                  // A[node][k0]
        a.y = arow[kc * 4 + 1];                  // A[node][k0+1]
        v2f b1, b2;
        b1.x = w1l[(k0 + 0) * FDIM];
        b1.y = w1l[(k0 + 1) * FDIM];
        b2.x = w2l[(k0 + 0) * FDIM];
        b2.y = w2l[(k0 + 1) * FDIM];
        c1 = __builtin_amdgcn_wmma_f32_16x16x4_f32(false, a, false, b1, (short)0, c1, false, false);
        c2 = __builtin_amdgcn_wmma_f32_16x16x4_f32(false, a, false, b2, (short)0, c2, false, false);
    }

#pragma unroll
    for (int r = 0; r < 8; ++r) {
        const int node = node0 + r + half * 8;
        const float inv_d = 1.0f / deg[node];
        const float h1 = c1[r] * inv_d;
        const float h2 = c2[r] * inv_d;
        const float o1 = h1 > 0.0f ? h1 : 0.0f;
        const float o2 = 1.0f / (1.0f + __expf(-h2));
        out[node * FDIM + nBase + l16] = o1 * o2;
    }
}

// ---------------- launcher ----------------

extern "C" void kernel_launch(void* const* d_in, const int* in_sizes, int n_in,
                              void* d_out, int out_size, void* d_ws, size_t ws_size,
                              hipStream_t stream) {
    const float*     x   = (const float*)d_in[0];
    const long long* ei  = (const long long*)d_in[1];
    const float*     W1  = (const float*)d_in[2];
    const float*     W2  = (const float*)d_in[3];
    const int E = in_sizes[1] / 2;                 // 1,600,000
    const long long* srcs = ei;
    const long long* dsts = ei + E;

    float* deg  = (float*)d_ws;                    // [N]
    float* dinv = deg + N_NODES;                   // [N]
    float* A    = dinv + N_NODES;                  // [N,64]
    float* out  = (float*)d_out;

    const int nB  = (N_NODES + 255) / 256;
    const int eB  = (E + 255) / 256;
    const int aB  = (N_NODES * FDIM + 255) / 256;
    const int esB = (int)(((long long)E * 32 + 255) / 256);

    init_deg_kernel<<<nB, 256, 0, stream>>>(deg);
    count_deg_kernel<<<eB, 256, 0, stream>>>(dsts, deg, E);
    dinv_kernel<<<nB, 256, 0, stream>>>(deg, dinv);
    agg_init_kernel<<<aB, 256, 0, stream>>>(x, dinv, A);
    edge_scatter_kernel<<<esB, 256, 0, stream>>>(srcs, dsts, x, dinv, A, E);
    gcn_wmma_kernel<<<N_NODES / 16, 128, 0, stream>>>(A, W1, W2, deg, out);
}